// SelfAttention_61572651155538
// MI455X (gfx1250) — compile-verified
//
#include <hip/hip_runtime.h>

// Problem constants (from reference setup_inputs)
#define BB   4
#define TQ   512
#define TK   512
#define DQK  512   // DQ == DV == 512
#define UU   256

typedef float v2f __attribute__((ext_vector_type(2)));
typedef float v8f __attribute__((ext_vector_type(8)));

// ---------------- fast transcendental helpers (gfx1250) ----------------
__device__ __forceinline__ float fast_exp(float x) {
#if __has_builtin(__builtin_amdgcn_exp2f)
    return __builtin_amdgcn_exp2f(x * 1.4426950408889634f);
#else
    return __expf(x);
#endif
}

__device__ __forceinline__ float fast_rcp(float x) {
#if __has_builtin(__builtin_amdgcn_rcpf)
    return __builtin_amdgcn_rcpf(x);
#else
    return 1.0f / x;
#endif
}

__device__ __forceinline__ float fast_tanh(float x) {
#if __has_builtin(__builtin_amdgcn_tanhf)
    return __builtin_amdgcn_tanhf(x);          // v_tanh_f32 (gfx1250 trans op)
#else
    // tanh(x) = 1 - 2/(e^{2x}+1); exp2/rcp are single trans ops
#if __has_builtin(__builtin_amdgcn_exp2f)
    float t = __builtin_amdgcn_exp2f(x * 2.8853900817779268f); // 2*log2(e)
#else
    float t = __expf(2.0f * x);
#endif
    return 1.0f - 2.0f * fast_rcp(t + 1.0f);
#endif
}

// ---------------------------------------------------------------------------
// Kernel 1: f32 WMMA projection  Out[row, col] = sum_d A[row,d] * W[d,col]
//   A: [nRows, DQK] row-major, W: [DQK, UU] row-major.
//   One wave per 16x16 output tile; K-loop in steps of 4 (V_WMMA_F32_16X16X4_F32).
//   TRANSPOSE=true stores Out as Kt[b][u][k]  (b = row/TK, k = row%TK, u = col).
// ---------------------------------------------------------------------------
template <bool TRANSPOSE>
__global__ __launch_bounds__(256) void proj_wmma_f32(const float* __restrict__ A,
                                                     const float* __restrict__ W,
                                                     float* __restrict__ Out)
{
    const int lane = threadIdx.x & 31;
    const int wave = threadIdx.x >> 5;
    const int tile = blockIdx.x * 8 + wave;

    const int colTiles = UU / 16;
    const int rowBase  = (tile / colTiles) * 16;
    const int colBase  = (tile % colTiles) * 16;

    const int l15 = lane & 15;
    const int hi  = lane >> 4;                 // 0 for lanes 0-15, 1 for 16-31

    // A fragment: lane (hi,l15) supplies A[rowBase+l15, d + 2*hi + {0,1}]
    const float* arow = A + (size_t)(rowBase + l15) * DQK;
    // B fragment: lane supplies W[d + hi + {0,2}, colBase + l15]
    const int col = colBase + l15;

    v8f c = {};
    for (int d = 0; d < DQK; d += 4) {
        v2f a, b;
        a.x = arow[d + 2 * hi + 0];
        a.y = arow[d + 2 * hi + 1];
        b.x = W[(size_t)(d + hi + 0) * UU + col];
        b.y = W[(size_t)(d + hi + 2) * UU + col];
        c = __builtin_amdgcn_wmma_f32_16x16x4_f32(
                /*neg_a=*/false, a, /*neg_b=*/false, b,
                /*c_mod=*/(short)0, c, /*reuse_a=*/false, /*reuse_b=*/false);
    }

    // C/D layout: VGPR r -> M = r + 8*hi, N = l15
#pragma unroll
    for (int r = 0; r < 8; ++r) {
        const int row = rowBase + r + 8 * hi;
        if (TRANSPOSE) {
            const int b_ = row / TK;
            const int k_ = row % TK;
            Out[((size_t)b_ * UU + col) * TK + k_] = c[r];
        } else {
            Out[(size_t)row * UU + col] = c[r];
        }
    }
}

// ---------------------------------------------------------------------------
// Kernel 2: fused additive-attention scores + row softmax.
//   Grid: BB * (TQ/QTILE) workgroups, 512 threads (16 wave32).
//   Thread tid owns column k = tid for QTILE query rows.
// ---------------------------------------------------------------------------
#define QTILE 8
#define BLK   512

__global__ __launch_bounds__(BLK) void addattn_scores_softmax(
        const float* __restrict__ Qp,    // [BB, TQ, UU]
        const float* __restrict__ Kt,    // [BB, UU, TK]  (transposed)
        const float* __restrict__ scale, // [UU]
        float* __restrict__ out)         // [BB, TQ, TK]
{
    __shared__ float qv[QTILE][UU];
    __shared__ float ssc[UU];
    __shared__ float red[16];

    const int tid  = threadIdx.x;
    const int lane = tid & 31;
    const int wid  = tid >> 5;
    const int b    = blockIdx.x / (TQ / QTILE);
    const int q0   = (blockIdx.x % (TQ / QTILE)) * QTILE;

    // Stage query tile + scale in LDS (coalesced)
    for (int i = tid; i < QTILE * UU; i += BLK) {
        const int qi = i / UU, u = i % UU;
        qv[qi][u] = Qp[((size_t)b * TQ + q0 + qi) * UU + u];
    }
    for (int i = tid; i < UU; i += BLK) ssc[i] = scale[i];
    __syncthreads();

    const float* ktb = Kt + (size_t)b * UU * TK + tid;   // Kt[b][u][tid], u-stride TK

    float acc[QTILE];
#pragma unroll
    for (int qi = 0; qi < QTILE; ++qi) acc[qi] = 0.0f;

    // Hot loop: 1 coalesced load + QTILE * (LDS bcast + tanh + fma) per u
#pragma unroll 4
    for (int u = 0; u < UU; ++u) {
        const float kv = ktb[(size_t)u * TK];
        const float s  = ssc[u];
#pragma unroll
        for (int qi = 0; qi < QTILE; ++qi) {
            acc[qi] = fmaf(s, fast_tanh(qv[qi][u] + kv), acc[qi]);
        }
    }

    // Row softmax over k (512 threads) for each of the QTILE rows
    for (int qi = 0; qi < QTILE; ++qi) {
        const float v = acc[qi];

        // --- max reduction ---
        float m = v;
#pragma unroll
        for (int off = 16; off >= 1; off >>= 1)
            m = fmaxf(m, __shfl_xor(m, off, 32));
        if (lane == 0) red[wid] = m;
        __syncthreads();
        m = red[0];
#pragma unroll
        for (int w = 1; w < 16; ++w) m = fmaxf(m, red[w]);
        __syncthreads();                       // protect red[] before reuse

        // --- exp + sum reduction ---
        const float e = fast_exp(v - m);
        float s = e;
#pragma unroll
        for (int off = 16; off >= 1; off >>= 1)
            s += __shfl_xor(s, off, 32);
        if (lane == 0) red[wid] = s;
        __syncthreads();
        float tot = 0.0f;
#pragma unroll
        for (int w = 0; w < 16; ++w) tot += red[w];
        __syncthreads();

        out[((size_t)b * TQ + q0 + qi) * TK + tid] = e * fast_rcp(tot);
    }
}

// ---------------------------------------------------------------------------
extern "C" void kernel_launch(void* const* d_in, const int* in_sizes, int n_in,
                              void* d_out, int out_size, void* d_ws, size_t ws_size,
                              hipStream_t stream)
{
    const float* query = (const float*)d_in[0];   // [BB, TQ, DQK]
    const float* value = (const float*)d_in[1];   // [BB, TK, DQK]
    const float* W1    = (const float*)d_in[2];   // [DQK, UU]
    const float* W2    = (const float*)d_in[3];   // [DQK, UU]
    const float* scale = (const float*)d_in[4];   // [UU]
    float*       out   = (float*)d_out;           // [BB, TQ, TK]

    // Workspace: Qp (2 MB) + Kt (2 MB)
    float* Qp = (float*)d_ws;
    float* Kt = Qp + (size_t)BB * TQ * UU;

    // 2048 rows x 256 cols -> 2048 16x16 tiles -> 256 blocks of 8 waves each
    const int tiles  = (BB * TQ / 16) * (UU / 16);
    const int blocks = tiles / 8;

    proj_wmma_f32<false><<<blocks, 256, 0, stream>>>(query, W1, Qp);
    proj_wmma_f32<true ><<<blocks, 256, 0, stream>>>(value, W2, Kt);

    addattn_scores_softmax<<<BB * (TQ / QTILE), BLK, 0, stream>>>(Qp, Kt, scale, out);
}